// QuantizationLayer_45715631898659
// MI455X (gfx1250) — compile-verified
//
#include <hip/hip_runtime.h>
#include <stdint.h>

// ---------------------------------------------------------------------------
// QuantizationLayer: out[n, d*4 + k] = bit (3-k) of round(x[n,d]*16 - 0.5)
// Pure streaming problem: 128 MB in + 512 MB out => HBM-bound (~27.5us floor
// @ 23.3 TB/s). No matmul structure => WMMA inapplicable. Use CDNA5 async
// global->LDS pipeline + NT cache policy for the data movement path.
// ---------------------------------------------------------------------------

typedef float v4f __attribute__((ext_vector_type(4)));

#if defined(__has_builtin)
#if __has_builtin(__builtin_amdgcn_global_load_async_to_lds_b128) && \
    __has_builtin(__builtin_amdgcn_s_wait_asynccnt)
#define USE_ASYNC 1
#endif
#endif
#ifndef USE_ASYNC
#define USE_ASYNC 0
#endif

#if USE_ASYNC
// Builtin expects: (global int4-vec*, lds int4-vec*, imm offset, imm cpol)
// (clang diagnostic: "__attribute__((__vector_size__(4*sizeof(int)))) int
//  __device__ *" for arg0 — i.e. AS1 pointer to a 16-byte int vector)
typedef int async_b128_t __attribute__((vector_size(16)));
typedef __attribute__((address_space(1))) async_b128_t* async_gptr_t;
typedef __attribute__((address_space(3))) async_b128_t* async_lptr_t;

__device__ __forceinline__ void async_copy_b128(const float* gp, float* lp) {
  __builtin_amdgcn_global_load_async_to_lds_b128(
      (async_gptr_t)(uintptr_t)gp,  // strip const, addrspacecast to AS1
      (async_lptr_t)lp,             // addrspacecast generic->AS3
      /*offset=*/0, /*cpol=*/0);
}
#endif

__device__ __forceinline__ v4f unpack4_b4(float xv) {
  // jnp.round == round-half-to-even == v_rndne_f32 (rintf)
  int q = ((int)rintf(xv * 16.0f - 0.5f)) & 0xFF;
  v4f r;
  r.x = (float)((q >> 3) & 1);
  r.y = (float)((q >> 2) & 1);
  r.z = (float)((q >> 1) & 1);
  r.w = (float)(q & 1);
  return r;
}

// One kernel symbol; device pass selects the async-pipelined body when the
// gfx1250 async builtins exist, else a direct NT streaming body. Both work
// with the same <<<2048, 256>>> launch geometry.
__global__ void __launch_bounds__(256) quant_b4_kernel(
    const float* __restrict__ x, float* __restrict__ out, int n) {
  const int tid = (int)threadIdx.x;
#if USE_ASYNC
  // 2 x 4KB double-buffered LDS tiles (trivial vs 320KB/WGP -> max occupancy)
  __shared__ float lds[2][1024];
  const int numTiles = n >> 10;          // 1024 floats per tile
  const int gsz = (int)gridDim.x;

  if ((int)blockIdx.x < numTiles) {
    // Prologue: issue first tile into buffer 0 (one B128 async per lane).
    async_copy_b128(x + (((size_t)blockIdx.x) << 10) + (size_t)(tid << 2),
                    &lds[0][tid << 2]);
    int buf = 0;
    for (int tile = (int)blockIdx.x; tile < numTiles; tile += gsz) {
      const int nextTile = tile + gsz;
      if (nextTile < numTiles) {
        // Issue next tile into the other buffer, then wait for current only.
        async_copy_b128(x + (((size_t)nextTile) << 10) + (size_t)(tid << 2),
                        &lds[buf ^ 1][tid << 2]);
        __builtin_amdgcn_s_wait_asynccnt(1);  // async loads complete in order
      } else {
        __builtin_amdgcn_s_wait_asynccnt(0);
      }
      __syncthreads();  // all waves' async data for `buf` is now in LDS

      const size_t base = ((size_t)tile) << 10;  // in input elements
      v4f* o4 = (v4f*)out;                       // out float4 index == in index
#pragma unroll
      for (int s = 0; s < 4; ++s) {
        const int l = (s << 8) + tid;            // conflict-free LDS read
        v4f r = unpack4_b4(lds[buf][l]);
        // coalesced 512B/wave store; NT: don't pollute L2 with dead stream
        __builtin_nontemporal_store(r, o4 + base + (size_t)l);
      }
      __syncthreads();  // buffer free before it is re-filled next iteration
      buf ^= 1;
    }
  }
  // Remainder if n % 1024 != 0 (not hit for 65536*512, kept for safety)
  const int remStart = numTiles << 10;
  for (int i = remStart + (int)blockIdx.x * 256 + tid; i < n;
       i += gsz * 256) {
    v4f r = unpack4_b4(x[i]);
    __builtin_nontemporal_store(r, (v4f*)out + (size_t)i);
  }
#else
  // Fallback: direct NT streaming, 1 float in -> 1 float4 out per thread,
  // fully coalesced both directions.
  const size_t stride = (size_t)gridDim.x * (size_t)blockDim.x;
  for (size_t i = (size_t)blockIdx.x * blockDim.x + (size_t)tid; i < (size_t)n;
       i += stride) {
    float v = __builtin_nontemporal_load(x + i);
    __builtin_nontemporal_store(unpack4_b4(v), (v4f*)out + i);
  }
#endif
}

// Generic-B scalar fallback (reference fixes B=4; kept for robustness).
__global__ void __launch_bounds__(256) quant_generic_kernel(
    const float* __restrict__ x, float* __restrict__ out, long long total,
    int Bv) {
  const float step = (float)(1 << Bv);
  const long long stride = (long long)gridDim.x * blockDim.x;
  for (long long o = (long long)blockIdx.x * blockDim.x + threadIdx.x;
       o < total; o += stride) {
    long long i = o / Bv;
    int k = (int)(o - i * (long long)Bv);
    int q = ((int)rintf(x[i] * step - 0.5f)) & 0xFF;
    out[o] = (float)((q >> (Bv - 1 - k)) & 1);
  }
}

extern "C" void kernel_launch(void* const* d_in, const int* in_sizes, int n_in,
                              void* d_out, int out_size, void* d_ws,
                              size_t ws_size, hipStream_t stream) {
  const float* x = (const float*)d_in[0];
  float* out = (float*)d_out;
  const long long n = (long long)in_sizes[0];
  const int Bv = (n > 0) ? (int)((long long)out_size / n) : 0;

  if (Bv == 4) {
    // 2048 blocks * 8 waves = 16K waves in flight; each block walks tiles
    // grid-stride with a 2-deep async pipeline.
    quant_b4_kernel<<<2048, 256, 0, stream>>>(x, out, (int)n);
  } else {
    quant_generic_kernel<<<4096, 256, 0, stream>>>(x, out, (long long)out_size,
                                                   Bv);
  }
}